// LinformerProjectionEntireOutImg_10479720202690
// MI455X (gfx1250) — compile-verified
//
#include <hip/hip_runtime.h>
#include <hip/hip_bf16.h>

typedef __attribute__((ext_vector_type(2))) float v2f;
typedef __attribute__((ext_vector_type(8))) float v8f;
typedef __attribute__((ext_vector_type(2))) int   v2i;

#if defined(__gfx1250__) && __has_builtin(__builtin_amdgcn_global_load_async_to_lds_b64)
#define HAS_ASYNC_LDS 1
typedef __attribute__((address_space(1))) v2i* gptr_v2i;  // global (prints as __device__)
typedef __attribute__((address_space(3))) v2i* lptr_v2i;  // LDS
#else
#define HAS_ASYNC_LDS 0
#endif

#if __has_builtin(__builtin_amdgcn_s_wait_asynccnt)
#define WAIT_ASYNC(n) __builtin_amdgcn_s_wait_asynccnt(n)
#else
#define WAIT_ASYNC(n) asm volatile("s_wait_asynccnt %0" ::"i"(n) : "memory")
#endif

// ---------------------------------------------------------------------------
// Kernel A: G[b][m*64+k] = sum_pose ( A[b, pose*128+m] (8x8) @ W[pose*128+m] )
// One block = (b, tile of 8 consecutive m). 256 threads, 8 waves (wave32).
// Double-buffered GLOBAL_LOAD_ASYNC_TO_LDS_B64 pipeline (ASYNCcnt): issue
// pose+1's 2KB A / 2KB W tiles into buf^1, s_wait_asynccnt 2 to ensure pose's
// tiles landed, barrier, compute. Deterministic per-thread reduction.
// ---------------------------------------------------------------------------
__global__ __launch_bounds__(256) void reduce_pose_kernel(
    const float* __restrict__ cp, const float* __restrict__ wc,
    float* __restrict__ G) {
  __shared__ float As[2][512];
  __shared__ float Ws[2][512];
  const int b   = blockIdx.y;
  const int m0  = blockIdx.x * 8;          // 16 tiles of 8 m's
  const int tid = threadIdx.x;
  const int e0  = tid * 2;                 // two outputs per thread (same row)
  const int mm  = e0 >> 6;                 // which of the 8 matrices
  const int rr  = (e0 >> 3) & 7;           // row
  const int cc  = e0 & 7;                  // col (even), cc+1 is second output
  float acc0 = 0.f, acc1 = 0.f;
  const float* Abase = cp + ((size_t)b * 8192 + (size_t)m0) * 64;
  const float* Wbase = wc + (size_t)m0 * 64;

#if HAS_ASYNC_LDS
  // ---- async-to-LDS double-buffered pipeline ----
#define ISSUE_POSE(pose, buf)                                                  \
  do {                                                                         \
    float* Ap_ = const_cast<float*>(Abase + (size_t)(pose) * (128 * 64) + e0); \
    float* Wp_ = const_cast<float*>(Wbase + (size_t)(pose) * (128 * 64) + e0); \
    __builtin_amdgcn_global_load_async_to_lds_b64(                             \
        (gptr_v2i)(void*)Ap_, (lptr_v2i)(void*)&As[buf][e0], 0, 0);            \
    __builtin_amdgcn_global_load_async_to_lds_b64(                             \
        (gptr_v2i)(void*)Wp_, (lptr_v2i)(void*)&Ws[buf][e0], 0, 0);            \
  } while (0)

  ISSUE_POSE(0, 0);
  for (int pose = 0; pose < 64; ++pose) {
    const int buf = pose & 1;
    if (pose + 1 < 64) {
      ISSUE_POSE(pose + 1, buf ^ 1);
      WAIT_ASYNC(2);   // 2 newest (pose+1) may stay in flight; pose's are done
    } else {
      WAIT_ASYNC(0);
    }
    __syncthreads();
    const float* arow = &As[buf][mm * 64 + rr * 8];
    const float* wcol = &Ws[buf][mm * 64 + cc];
#pragma unroll
    for (int t = 0; t < 8; ++t) {
      const float av = arow[t];
      acc0 = fmaf(av, wcol[t * 8 + 0], acc0);
      acc1 = fmaf(av, wcol[t * 8 + 1], acc1);
    }
    __syncthreads();   // LDS buf is refilled by the next iteration's issue
  }
#undef ISSUE_POSE
#else
  // ---- fallback: manual VGPR staging ----
  for (int pose = 0; pose < 64; ++pose) {
    const float* Ap = Abase + (size_t)pose * (128 * 64);
    const float* Wp = Wbase + (size_t)pose * (128 * 64);
    *(float2*)&As[0][e0] = *(const float2*)&Ap[e0];
    *(float2*)&Ws[0][e0] = *(const float2*)&Wp[e0];
    __builtin_prefetch(Ap + 128 * 64 + e0, 0, 0);
    __syncthreads();
    const float* arow = &As[0][mm * 64 + rr * 8];
    const float* wcol = &Ws[0][mm * 64 + cc];
#pragma unroll
    for (int t = 0; t < 8; ++t) {
      const float av = arow[t];
      acc0 = fmaf(av, wcol[t * 8 + 0], acc0);
      acc1 = fmaf(av, wcol[t * 8 + 1], acc1);
    }
    __syncthreads();
  }
#endif

  float2 r; r.x = acc0; r.y = acc1;
  *(float2*)&G[(size_t)b * 8192 + (size_t)m0 * 64 + e0] = r;
}

// ---------------------------------------------------------------------------
// Kernel B: F[rs][c] = sum_{j=0..3} E_proj[rs*256 + c + 64*j]   (rs = nh*256+s2)
// ---------------------------------------------------------------------------
__global__ __launch_bounds__(256) void fold_eproj_kernel(
    const float* __restrict__ E, float* __restrict__ F) {
  const int idx = blockIdx.x * 256 + threadIdx.x;  // 524288 outputs
  const int rs = idx >> 6;
  const int c  = idx & 63;
  const float* p = E + (size_t)rs * 256 + c;
  F[idx] = (p[0] + p[64]) + (p[128] + p[192]);
}

// ---------------------------------------------------------------------------
// Kernel C: split-K f32 WMMA GEMM  pooled_partial = G(32x8192) @ F(8192x64).
// 32 blocks, each owns K-chunk of 256. 8 waves/block, each wave one 16x16
// C tile (2 M-tiles x 4 N-tiles). Uses V_WMMA_F32_16X16X4_F32.
//   A 16x4 lane layout: lane<16 -> M=lane, K={k+0,k+1}; lane>=16 -> K={k+2,k+3}
//   B 4x16 lane layout (symmetric): VGPR0 = B[k+2*(lane/16)][lane%16], VGPR1 = +1
// ---------------------------------------------------------------------------
__global__ __launch_bounds__(256) void pooled_gemm_kernel(
    const float* __restrict__ G, const float* __restrict__ F,
    float* __restrict__ partial) {
  const int chunk = blockIdx.x;            // K range [chunk*256, +256)
  const int wave  = threadIdx.x >> 5;
  const int lane  = threadIdx.x & 31;
  const int tileM = wave >> 2;             // 0..1
  const int tileN = wave & 3;              // 0..3
  const int half  = lane >> 4;             // 0/1
  const int l16   = lane & 15;
  const int arow  = tileM * 16 + l16;      // A-matrix row for this lane
  const int bcol  = tileN * 16 + l16;      // B-matrix column for this lane
  v8f c = {0.f, 0.f, 0.f, 0.f, 0.f, 0.f, 0.f, 0.f};
  const int k0 = chunk * 256;
  const float* Ga = G + (size_t)arow * 8192 + k0 + 2 * half;
  const float* Fb = F + (size_t)(k0 + 2 * half) * 64 + bcol;
#pragma unroll 4
  for (int kk = 0; kk < 256; kk += 4) {
    v2f a = *(const v2f*)(Ga + kk);        // G[arow][k + 2*half + {0,1}]
    v2f b;
    b.x = Fb[(size_t)kk * 64];             // F[k + 2*half + 0][bcol]
    b.y = Fb[(size_t)kk * 64 + 64];        // F[k + 2*half + 1][bcol]
    c = __builtin_amdgcn_wmma_f32_16x16x4_f32(
        /*neg_a=*/false, a, /*neg_b=*/false, b,
        /*c_mod=*/(short)0, c, /*reuse_a=*/false, /*reuse_b=*/false);
  }
  // C/D layout: VGPR i holds row tileM*16 + 8*half + i, col bcol
#pragma unroll
  for (int i = 0; i < 8; ++i) {
    const int r = tileM * 16 + half * 8 + i;
    partial[(size_t)chunk * 2048 + (size_t)r * 64 + bcol] = c[i];
  }
}

// ---------------------------------------------------------------------------
// Kernel D: deterministic split-K reduce, add rel_embedd, then per-(b,o)
// 8x8 @ w_next[o]. One block per b (64 threads, 2 waves), thread t owns o=t.
// ---------------------------------------------------------------------------
__global__ __launch_bounds__(64) void finalize_kernel(
    const float* __restrict__ partial, const float* __restrict__ rel,
    const float* __restrict__ w_next, float* __restrict__ out) {
  __shared__ float npc[64];
  const int b = blockIdx.x;
  const int t = threadIdx.x;               // 0..63: pooled col, then o index
  float s = 0.f;
#pragma unroll
  for (int ch = 0; ch < 32; ++ch) s += partial[ch * 2048 + b * 64 + t];
  npc[t] = s * (1.0f / 64.0f) + rel[t];    // pooled/OUT_N + rel_embedd
  __syncthreads();
  const float* W = w_next + t * 64;        // w_next[o=t] 8x8 row-major
  float* dst = out + (size_t)b * 4096 + (size_t)t * 64;
#pragma unroll
  for (int r = 0; r < 8; ++r) {
    float a[8];
#pragma unroll
    for (int q = 0; q < 8; ++q) a[q] = npc[r * 8 + q];
#pragma unroll
    for (int ccx = 0; ccx < 8; ++ccx) {
      float v = 0.f;
#pragma unroll
      for (int q = 0; q < 8; ++q) v = fmaf(a[q], W[q * 8 + ccx], v);
      dst[r * 8 + ccx] = v;
    }
  }
}

extern "C" void kernel_launch(void* const* d_in, const int* in_sizes, int n_in,
                              void* d_out, int out_size, void* d_ws, size_t ws_size,
                              hipStream_t stream) {
  const float* current_pose = (const float*)d_in[0];  // (32, 8192, 64)
  const float* w_current    = (const float*)d_in[1];  // (1,1,8192,8,8)
  const float* w_next       = (const float*)d_in[2];  // (64,8,8)
  const float* E_proj       = (const float*)d_in[3];  // (32,256,256)
  const float* rel_embedd   = (const float*)d_in[4];  // (1,1,64)
  float* out = (float*)d_out;                         // (32,1,64,64)

  float* G       = (float*)d_ws;        // 32*8192        = 262144 f32
  float* F       = G + 262144;          // 8192*64        = 524288 f32
  float* partial = F + 524288;          // 32 chunks*32*64 =  65536 f32
  (void)in_sizes; (void)n_in; (void)out_size; (void)ws_size;

  reduce_pose_kernel<<<dim3(16, 32), 256, 0, stream>>>(current_pose, w_current, G);
  fold_eproj_kernel<<<2048, 256, 0, stream>>>(E_proj, F);
  pooled_gemm_kernel<<<32, 256, 0, stream>>>(G, F, partial);
  finalize_kernel<<<32, 64, 0, stream>>>(partial, rel_embedd, w_next, out);
}